// Loss_88321707475228
// MI455X (gfx1250) — compile-verified
//
#include <hip/hip_runtime.h>
#include <hip/hip_bf16.h>

#define BATCH 8
#define CHN   80
#define HH    128
#define WW    128
#define HWSZ  (HH * WW)            // 16384
#define KK    32
#define RES_F 1e-4f
#define PULL_W 0.1f
#define PUSH_W 0.1f
#define OFF_W  1.0f

// workspace layout (floats)
#define WS_TL_LOSS 0                 // [8]
#define WS_TL_NP   8                 // [8]
#define WS_BR_LOSS 16                // [8]
#define WS_BR_NP   24                // [8]
#define WS_TL_OFF  32                // [8]
#define WS_BR_OFF  40                // [8]
#define WS_TL_LIST 48                // [8*33]
#define WS_BR_LIST 312               // [8*33]
#define WS_N       576

typedef __attribute__((ext_vector_type(2))) float v2f;
typedef __attribute__((ext_vector_type(8))) float v8f;
typedef __attribute__((ext_vector_type(4))) float v4f;   // native vector: OK for
typedef __attribute__((ext_vector_type(4))) int   v4i;   // __builtin_nontemporal_load

// ---------------------------------------------------------------------------
// zero the workspace accumulators (must run first every launch)
// ---------------------------------------------------------------------------
__global__ void zero_ws_kernel(float* __restrict__ ws, int n) {
    int i = blockIdx.x * blockDim.x + threadIdx.x;
    if (i < n) ws[i] = 0.0f;
}

// ---------------------------------------------------------------------------
// block reduction helper: 256 threads, two values, atomic into global
// ---------------------------------------------------------------------------
__device__ __forceinline__ void block_reduce2(float v1, float v2,
                                              float* o1, float* o2) {
    __shared__ float s1[256];
    __shared__ float s2[256];
    int t = threadIdx.x;
    s1[t] = v1; s2[t] = v2;
    __syncthreads();
    for (int off = 128; off > 0; off >>= 1) {
        if (t < off) { s1[t] += s1[t + off]; s2[t] += s2[t + off]; }
        __syncthreads();
    }
    if (t == 0) {
        atomicAdd(o1, s1[0]);
        if (o2) atomicAdd(o2, s2[0]);
    }
}

// ---------------------------------------------------------------------------
// focal loss: stream logits+gt with non-temporal b128 loads (read-once,
// 168 MB total -> don't thrash the 192 MB L2), accumulate (pos+neg) loss and
// num_pos per batch. grid = (256, BATCH), block = 256
// ---------------------------------------------------------------------------
__global__ void focal_kernel(const float* __restrict__ logits,
                             const float* __restrict__ gt,
                             float* __restrict__ ws_loss,
                             float* __restrict__ ws_np) {
    const int b = blockIdx.y;
    const size_t slab = (size_t)b * CHN * HWSZ;
    const v4f* lp = (const v4f*)(logits + slab);
    const v4f* gp = (const v4f*)(gt + slab);
    const int nvec = (CHN * HWSZ) / 4;   // 327680

    float loss = 0.0f, np = 0.0f;
    for (int i = blockIdx.x * blockDim.x + threadIdx.x; i < nvec;
         i += gridDim.x * blockDim.x) {
        v4f x4 = __builtin_nontemporal_load(&lp[i]);
        v4f g4 = __builtin_nontemporal_load(&gp[i]);
#pragma unroll
        for (int c = 0; c < 4; ++c) {
            float x = x4[c], g = g4[c];
            float s = 1.0f / (1.0f + __expf(-x));               // sigmoid
            float p = fminf(fmaxf(s, RES_F), 1.0f - RES_F);     // clip
            bool  pos = (g == 1.0f);
            float q  = 1.0f - p;
            float t  = 1.0f - g;
            float t4 = (t * t) * (t * t);
            float larg  = pos ? p : (1.0f - p);                 // single log
            float coeff = pos ? (q * q) : (t4 * p * p);
            loss += coeff * __logf(larg);
            np   += pos ? 1.0f : 0.0f;
        }
    }
    block_reduce2(loss, np, &ws_loss[b], &ws_np[b]);
}

// ---------------------------------------------------------------------------
// offset (smooth-L1) loss, masked by pos!=0 broadcast over the 2 channels.
// grid = (8, BATCH), block = 256
// ---------------------------------------------------------------------------
__global__ void offset_kernel(const float* __restrict__ pred,
                              const float* __restrict__ gtp,
                              const int*  __restrict__ pos,
                              float* __restrict__ ws_out) {
    const int b = blockIdx.y;
    const v4f* pp = (const v4f*)(pred + (size_t)b * 2 * HWSZ);
    const v4f* gp = (const v4f*)(gtp + (size_t)b * 2 * HWSZ);
    const v4i* mp = (const v4i*)(pos + (size_t)b * HWSZ);
    const int nvec = (2 * HWSZ) / 4;      // 8192
    const int hv   = HWSZ / 4;            // 4096

    float acc = 0.0f;
    for (int i = blockIdx.x * blockDim.x + threadIdx.x; i < nvec;
         i += gridDim.x * blockDim.x) {
        v4f p4 = __builtin_nontemporal_load(&pp[i]);
        v4f g4 = __builtin_nontemporal_load(&gp[i]);
        v4i m4 = mp[i % hv];              // channel-broadcast mask (re-read)
#pragma unroll
        for (int c = 0; c < 4; ++c) {
            float d = fabsf(p4[c] - g4[c]);
            float l = (d < 1.0f) ? (0.5f * d * d) : (d - 0.5f);
            acc += (m4[c] != 0) ? l : 0.0f;
        }
    }
    block_reduce2(acc, 0.0f, &ws_out[b], nullptr);
}

// ---------------------------------------------------------------------------
// segment-sum gather of embeds into (B, K+1) lists (segment 0 skipped)
// ---------------------------------------------------------------------------
__global__ void gather_kernel(const float* __restrict__ tl_emb,
                              const int*  __restrict__ tl_pos,
                              const float* __restrict__ br_emb,
                              const int*  __restrict__ br_pos,
                              float* __restrict__ tl_list,
                              float* __restrict__ br_list) {
    int idx = blockIdx.x * blockDim.x + threadIdx.x;
    if (idx >= BATCH * HWSZ) return;
    int b = idx / HWSZ;
    int p1 = tl_pos[idx];
    if (p1 != 0) atomicAdd(&tl_list[b * (KK + 1) + p1], tl_emb[idx]);
    int p2 = br_pos[idx];
    if (p2 != 0) atomicAdd(&br_list[b * (KK + 1) + p2], br_emb[idx]);
}

// ---------------------------------------------------------------------------
// exact f32 wave32 reduction via V_WMMA_F32_16X16X4_F32 with B = ones:
//   A: a0 = x[lane], a1 = 0  ->  D[M][N] = x[M] + x[M+16]  (all N)
//   sum of D's 8 VGPRs per lane, then a second identical WMMA broadcasts
//   the full 32-lane sum to every lane.  EXEC must be all ones.
// ---------------------------------------------------------------------------
__device__ __forceinline__ float wave_reduce_wmma(float x) {
    v2f ones; ones.x = 1.0f; ones.y = 1.0f;
    v2f a;    a.x = x;       a.y = 0.0f;
    v8f c = {};
    c = __builtin_amdgcn_wmma_f32_16x16x4_f32(false, a, false, ones,
                                              (short)0, c, false, false);
    float t = c[0] + c[1] + c[2] + c[3] + c[4] + c[5] + c[6] + c[7];
    v2f a2;   a2.x = t;      a2.y = 0.0f;
    v8f c2 = {};
    c2 = __builtin_amdgcn_wmma_f32_16x16x4_f32(false, a2, false, ones,
                                               (short)0, c2, false, false);
    return c2[0];   // total on every lane
}

// ---------------------------------------------------------------------------
// finisher: single wave (32 threads). lane = embedding slot k-1 (k = 1..32).
// Computes pull/push with WMMA reductions, then combines focal + offset.
// ---------------------------------------------------------------------------
__global__ void final_kernel(const float* __restrict__ ws,
                             const int*  __restrict__ obj_mask,
                             float* __restrict__ out) {
    const int lane = threadIdx.x;   // 0..31, full wave, uniform flow
    const float* tl_loss = ws + WS_TL_LOSS;
    const float* tl_np   = ws + WS_TL_NP;
    const float* br_loss = ws + WS_BR_LOSS;
    const float* br_np   = ws + WS_BR_NP;
    const float* off_tl  = ws + WS_TL_OFF;
    const float* off_br  = ws + WS_BR_OFF;
    const float* tl_list = ws + WS_TL_LIST;
    const float* br_list = ws + WS_BR_LIST;

    float result = 0.0f;   // meaningful on lane 0
    for (int b = 0; b < BATCH; ++b) {
        float objf = (float)obj_mask[b];
        float tl = tl_list[b * (KK + 1) + 1 + lane];
        float br = br_list[b * (KK + 1) + 1 + lane];
        float mean = 0.5f * (tl + br);
        float dv   = tl - br;
        float pull_l = 0.5f * dv * dv;   // (tl-mean)^2 + (br-mean)^2

        // push: column sums of D = relu(1 - |m_i - m_j|), m = mean per lane
        float col = 0.0f;
        for (int i = 0; i < KK; ++i) {
            float mi = __shfl(mean, i, 32);
            col += fmaxf(0.0f, 1.0f - fabsf(mean - mi));
        }
        float sumD   = wave_reduce_wmma(col);     // full 32x32 sum
        float pull_b = wave_reduce_wmma(pull_l);  // 32-lane sum

        if (lane == 0) {
            result += PULL_W * pull_b / (objf + RES_F);
            result += PUSH_W * (sumD - objf) / (objf * (objf - 1.0f) + RES_F);
            result += -(tl_loss[b]) / tl_np[b] - (br_loss[b]) / br_np[b];
            result += OFF_W * (off_tl[b] + off_br[b]) / (objf + RES_F);
        }
    }
    if (lane == 0) out[0] = result / (float)BATCH;
}

// ---------------------------------------------------------------------------
extern "C" void kernel_launch(void* const* d_in, const int* in_sizes, int n_in,
                              void* d_out, int out_size, void* d_ws, size_t ws_size,
                              hipStream_t stream) {
    const float* tl_heats      = (const float*)d_in[0];
    const float* tl_embeds     = (const float*)d_in[1];
    const float* tl_offsets    = (const float*)d_in[2];
    const float* br_heats      = (const float*)d_in[3];
    const float* br_embeds     = (const float*)d_in[4];
    const float* br_offsets    = (const float*)d_in[5];
    const float* gt_tl_heats   = (const float*)d_in[6];
    const float* gt_br_heats   = (const float*)d_in[7];
    const float* gt_tl_offsets = (const float*)d_in[8];
    const float* gt_br_offsets = (const float*)d_in[9];
    const int*   gt_tl_pos     = (const int*)d_in[10];
    const int*   gt_br_pos     = (const int*)d_in[11];
    const int*   gt_obj_mask   = (const int*)d_in[12];
    float* ws  = (float*)d_ws;
    float* out = (float*)d_out;

    zero_ws_kernel<<<(WS_N + 255) / 256, 256, 0, stream>>>(ws, WS_N);

    dim3 fgrid(256, BATCH);
    focal_kernel<<<fgrid, 256, 0, stream>>>(tl_heats, gt_tl_heats,
                                            ws + WS_TL_LOSS, ws + WS_TL_NP);
    focal_kernel<<<fgrid, 256, 0, stream>>>(br_heats, gt_br_heats,
                                            ws + WS_BR_LOSS, ws + WS_BR_NP);

    dim3 ogrid(8, BATCH);
    offset_kernel<<<ogrid, 256, 0, stream>>>(tl_offsets, gt_tl_offsets,
                                             gt_tl_pos, ws + WS_TL_OFF);
    offset_kernel<<<ogrid, 256, 0, stream>>>(br_offsets, gt_br_offsets,
                                             gt_br_pos, ws + WS_BR_OFF);

    gather_kernel<<<(BATCH * HWSZ) / 256, 256, 0, stream>>>(
        tl_embeds, gt_tl_pos, br_embeds, gt_br_pos,
        ws + WS_TL_LIST, ws + WS_BR_LIST);

    final_kernel<<<1, 32, 0, stream>>>(ws, gt_obj_mask, out);
}